// MPSLayer_71244917506369
// MI455X (gfx1250) — compile-verified
//
#include <hip/hip_runtime.h>

typedef float v2f __attribute__((ext_vector_type(2)));
typedef float v8f __attribute__((ext_vector_type(8)));
typedef unsigned int u32x4 __attribute__((ext_vector_type(4)));
typedef int i32x4 __attribute__((ext_vector_type(4)));
typedef int i32x8 __attribute__((ext_vector_type(8)));
typedef int v4i __attribute__((vector_size(16)));   // matches async-builtin pointee type

#if __has_builtin(__builtin_amdgcn_tensor_load_to_lds)
#define HAVE_TDM 1
#else
#define HAVE_TDM 0
#endif
#if __has_builtin(__builtin_amdgcn_global_load_async_to_lds_b128)
#define HAVE_ASYNC_LDS 1
#else
#define HAVE_ASYNC_LDS 0
#endif

namespace {
constexpr int kBatch   = 2048;
constexpr int kSites   = 128;
constexpr int kF       = 64;
constexpr int kD       = 64;
constexpr int kMid     = 126;
constexpr int kStride  = 68;                 // padded LDS row stride (dwords): bank-conflict-free b64
constexpr int kRows16  = 16 * kStride;       // 1088 dwords: one padded 16x64 tile
constexpr int kXOff    = 0;                  // x tile for current site   [16][68]
constexpr int kVOff    = kRows16;            // running vector v          [16][68]
constexpr int kWOff    = 2 * kRows16;        // W double buffer starts here
constexpr int kDSlice  = 64 * kStride;       // 4352 dwords: W[d,:,:] slice, padded
constexpr int kWBuf    = 2 * kDSlice;        // 8704 dwords: 2 d-slices per buffer (DC=2)
constexpr int kLdsDw   = kWOff + 2 * kWBuf;  // 19584 dwords = 78336 bytes
}

__device__ __forceinline__ v8f wmma_f32(v2f a, v2f b, v8f c) {
  // D(16x16,f32) = A(16x4,f32) x B(4x16,f32) + C
  return __builtin_amdgcn_wmma_f32_16x16x4_f32(false, a, false, b, (short)0, c,
                                               false, false);
}

#if HAVE_ASYNC_LDS
__device__ __forceinline__ void async_cp16(const float* g, float* l) {
  // global -> LDS, 16B per lane, tracked by ASYNCcnt
  __builtin_amdgcn_global_load_async_to_lds_b128(
      (__attribute__((address_space(1))) v4i*)g,
      (__attribute__((address_space(3))) v4i*)l, 0, 0);
}
__device__ __forceinline__ void wait_async0() {
#if __has_builtin(__builtin_amdgcn_s_wait_asynccnt)
  __builtin_amdgcn_s_wait_asynccnt(0);
#else
  asm volatile("s_wait_asynccnt 0" ::: "memory");
#endif
}
#endif

#if HAVE_TDM
// One TDM descriptor copies `rows` rows of 64 dwords (global, contiguous) into
// LDS, inserting a 4-dword pad after every 64 dwords -> our stride-68 swizzle,
// done entirely by the Tensor Data Mover (TENSORcnt).
__device__ __forceinline__ void tdm_load_chunk(float* ldsDst, const float* gsrc,
                                               int rows) {
  unsigned lds_off = (unsigned)(unsigned long long)
      (__attribute__((address_space(3))) void*)ldsDst;
  unsigned long long ga = (unsigned long long)gsrc;
  // D# group0: count=1 | lds_addr | global_addr[56:0] | type=2 ("image")
  u32x4 g0 = { 1u, lds_off, (unsigned)ga,
               (unsigned)((ga >> 32) & 0x01FFFFFFull) | (2u << 30) };
  // D# group1: data_size=4B, pad_enable, pad_interval=64dw, pad_amount=4dw,
  //            tensor_dim0=64, tensor_dim1=rows, tile_dim0=64, tile_dim1=rows,
  //            tensor_dim0_stride=64
  i32x8 g1 = { (int)((2u << 16) | (1u << 20) | (5u << 22) | (3u << 25)),
               (int)(64u << 16),
               (int)((unsigned)rows << 16),
               (int)(64u << 16),
               rows,
               64, 0, 0 };
  i32x4 z4 = { 0, 0, 0, 0 };
#if __clang_major__ >= 23
  i32x8 z8 = { 0, 0, 0, 0, 0, 0, 0, 0 };
  __builtin_amdgcn_tensor_load_to_lds(g0, g1, z4, z4, z8, 0);
#else
  __builtin_amdgcn_tensor_load_to_lds(g0, g1, z4, z4, 0);
#endif
}
#endif

__launch_bounds__(128)
__global__ void mps_chain_kernel(const float* __restrict__ x,
                                 const float* __restrict__ w0,
                                 const float* __restrict__ w_mid,
                                 const float* __restrict__ w_last,
                                 float* __restrict__ out)
{
  extern __shared__ float smem[];
  const int t    = threadIdx.x;
  const int lane = t & 31;
  const int wv   = t >> 5;     // wave 0..3: owns e-quarter (mid sites) / k-quarter (end sites)
  const int bl   = lane & 15;  // batch row within 16-batch tile (A/C layout low index)
  const int hi   = lane >> 4;  // 0: K rows {0,1} / M 0..7 ; 1: K rows {2,3} / M 8..15
  const int b0   = blockIdx.x * 16;

  v8f acc[4];
  #pragma unroll
  for (int nt = 0; nt < 4; ++nt) {
    #pragma unroll
    for (int j = 0; j < 8; ++j) acc[nt][j] = 0.0f;
  }

  // ---- stage x[:, site, :] for this block's 16 batches into lds_x ----
  auto stage_x = [&](int site) {
    const int bb = t >> 3;
    const int e0 = (t & 7) * 8;
    const float* src = x + ((size_t)(b0 + bb) * kSites + site) * kF + e0;
    float* dst = &smem[kXOff + bb * kStride + e0];
#if HAVE_ASYNC_LDS
    async_cp16(src, dst);
    async_cp16(src + 4, dst + 4);
#else
    const float4* s4 = reinterpret_cast<const float4*>(src);
    float4 r0 = s4[0];
    float4 r1 = s4[1];
    float4* d4 = reinterpret_cast<float4*>(dst);
    d4[0] = r0;
    d4[1] = r1;
#endif
  };

  // ---- scatter this wave's partial C tiles (16x64) into LDS scratch at partOff ----
  auto store_partials = [&](int partOff) {
    #pragma unroll
    for (int nt = 0; nt < 4; ++nt) {
      #pragma unroll
      for (int j = 0; j < 8; ++j) {
        // C layout: element (m = j + 8*hi, n = 16*nt + bl)
        smem[partOff + wv * kRows16 + (j + 8 * hi) * kStride + 16 * nt + bl] = acc[nt][j];
      }
    }
  };

  // ---- sum 4 wave-partials -> lds_v (each thread owns 8 dwords) ----
  auto reduce_partials = [&](int partOff) {
    const int r  = t >> 3;
    const int c0 = (t & 7) * 8;
    float4 s0{}, s1{};
    #pragma unroll
    for (int w2 = 0; w2 < 4; ++w2) {
      const float4* p = reinterpret_cast<const float4*>(
          &smem[partOff + w2 * kRows16 + r * kStride + c0]);
      float4 a0 = p[0], a1 = p[1];
      s0.x += a0.x; s0.y += a0.y; s0.z += a0.z; s0.w += a0.w;
      s1.x += a1.x; s1.y += a1.y; s1.z += a1.z; s1.w += a1.w;
    }
    float4* d = reinterpret_cast<float4*>(&smem[kVOff + r * kStride + c0]);
    d[0] = s0;
    d[1] = s1;
  };

  // ---- inner WMMA work for one staged d-pair; this wave handles e in [16*wv, 16*wv+16) ----
  auto compute_phase = [&](int bufOff, int dBase) {
    #pragma unroll
    for (int dd = 0; dd < 2; ++dd) {
      const int d = dBase + dd;
      const float vv  = smem[kVOff + bl * kStride + d];   // lanes 16..31 broadcast lanes 0..15
      const int slice = bufOff + dd * kDSlice;
      #pragma unroll
      for (int e0i = 0; e0i < 4; ++e0i) {
        const int e0 = 16 * wv + 4 * e0i + 2 * hi;
        v2f xp = *reinterpret_cast<const v2f*>(&smem[kXOff + bl * kStride + e0]);
        v2f a  = xp * vv;                                  // A[b,(d,e)] = v[b,d]*x[b,e]
        #pragma unroll
        for (int nt = 0; nt < 4; ++nt) {
          v2f bv = *reinterpret_cast<const v2f*>(
              &smem[slice + (16 * nt + bl) * kStride + e0]); // B[(d,e),f] = W[d,f,e]
          acc[nt] = wmma_f32(a, bv, acc[nt]);
        }
      }
    }
  };

  // ================= prologue: v = x0 @ w0 (K = f, split across waves) =================
  {
    // stage w0^T (w0t[d][f] = w0[f][d]) into buf0, swizzled stride (transpose: VALU path)
    #pragma unroll
    for (int inst = 0; inst < 8; ++inst) {
      const int q = t * 4 + inst * 512;      // 4096 dwords total
      const int f = q >> 6;
      const int d = q & 63;
      float4 r = *reinterpret_cast<const float4*>(w0 + q);
      smem[kWOff + (d + 0) * kStride + f] = r.x;
      smem[kWOff + (d + 1) * kStride + f] = r.y;
      smem[kWOff + (d + 2) * kStride + f] = r.z;
      smem[kWOff + (d + 3) * kStride + f] = r.w;
    }
    stage_x(0);
#if HAVE_ASYNC_LDS
    wait_async0();
#endif
  }
  __syncthreads();
  #pragma unroll
  for (int ks = 0; ks < 4; ++ks) {
    const int f0 = 16 * wv + 4 * ks + 2 * hi;
    v2f a = *reinterpret_cast<const v2f*>(&smem[kXOff + bl * kStride + f0]);  // A = x0
    #pragma unroll
    for (int nt = 0; nt < 4; ++nt) {
      v2f bv = *reinterpret_cast<const v2f*>(&smem[kWOff + (16 * nt + bl) * kStride + f0]);
      acc[nt] = wmma_f32(a, bv, acc[nt]);   // partial v over this wave's f-quarter
    }
  }
  __syncthreads();   // all w0t reads done before buf0 is reused as partial scratch

  // ================= main chain: 126 mid sites =================
  for (int s = 0; s < kMid; ++s) {
    const float* wbase = w_mid + (size_t)s * (kD * kF * kD);

    // finalize previous site's partials -> lds_v ; stage x for this site
    store_partials(kWOff);               // buf0 is stale here
    #pragma unroll
    for (int nt = 0; nt < 4; ++nt)
      #pragma unroll
      for (int j = 0; j < 8; ++j) acc[nt][j] = 0.0f;
    __syncthreads();
    stage_x(s + 1);                      // mid site s consumes x[:, s+1, :]
    reduce_partials(kWOff);
#if HAVE_ASYNC_LDS
    wait_async0();
#endif
    __syncthreads();

    // stage d = {0,1} into buf0
#if HAVE_TDM
    if (wv == 0) {
      tdm_load_chunk(&smem[kWOff], wbase, 128);
      __builtin_amdgcn_s_wait_tensorcnt(0);
    }
#else
    #pragma unroll
    for (int inst = 0; inst < 16; ++inst) {
      const int q  = t * 4 + inst * 512;  // 8192 dwords
      const int ds = q >> 12;
      const int rm = q & 4095;
      const int f  = rm >> 6;
      const int e  = rm & 63;
      float4 r = *reinterpret_cast<const float4*>(wbase + q);
      *reinterpret_cast<float4*>(&smem[kWOff + ds * kDSlice + f * kStride + e]) = r;
    }
#endif
    __syncthreads();

    // 32 double-buffered phases, 2 bond indices (d) per phase
    for (int p = 0; p < 32; ++p) {
#if HAVE_TDM
      if (wv == 0 && p + 1 < 32)         // TDM prefetch of next d-pair while all waves compute
        tdm_load_chunk(&smem[kWOff + ((p + 1) & 1) * kWBuf],
                       wbase + (size_t)(2 * p + 2) * 4096, 128);
      compute_phase(kWOff + (p & 1) * kWBuf, 2 * p);
      if (wv == 0 && p + 1 < 32)
        __builtin_amdgcn_s_wait_tensorcnt(0);
#else
      float4 pref[16];
      if (p + 1 < 32) {
        const float4* src = reinterpret_cast<const float4*>(wbase + (size_t)(2 * p + 2) * 4096);
        #pragma unroll
        for (int inst = 0; inst < 16; ++inst) pref[inst] = src[t + inst * 128];
      }
      compute_phase(kWOff + (p & 1) * kWBuf, 2 * p);
      if (p + 1 < 32) {
        const int dst = kWOff + ((p + 1) & 1) * kWBuf;
        #pragma unroll
        for (int inst = 0; inst < 16; ++inst) {
          const int q  = t * 4 + inst * 512;
          const int ds = q >> 12;
          const int rm = q & 4095;
          const int f  = rm >> 6;
          const int e  = rm & 63;
          *reinterpret_cast<float4*>(&smem[dst + ds * kDSlice + f * kStride + e]) = pref[inst];
        }
      }
#endif
      __syncthreads();
    }
  }

  // ================= epilogue: u = v @ w_last ; out = sum_f u*x_last =================
  store_partials(kWOff);                 // finalize site 125 -> v
  #pragma unroll
  for (int nt = 0; nt < 4; ++nt)
    #pragma unroll
    for (int j = 0; j < 8; ++j) acc[nt][j] = 0.0f;
  __syncthreads();
  stage_x(kSites - 1);                   // x[:, 127, :]
  reduce_partials(kWOff);
#if HAVE_ASYNC_LDS
  wait_async0();
#endif
  __syncthreads();

  // stage w_last^T (wlt[f][d] = w_last[d][f]) into buf0
  #pragma unroll
  for (int inst = 0; inst < 8; ++inst) {
    const int q = t * 4 + inst * 512;
    const int d = q >> 6;
    const int f = q & 63;
    float4 r = *reinterpret_cast<const float4*>(w_last + q);
    smem[kWOff + (f + 0) * kStride + d] = r.x;
    smem[kWOff + (f + 1) * kStride + d] = r.y;
    smem[kWOff + (f + 2) * kStride + d] = r.z;
    smem[kWOff + (f + 3) * kStride + d] = r.w;
  }
  __syncthreads();

  // u[b,f] = sum_d v[b,d] * w_last[d,f]  (K = d, split across waves)
  #pragma unroll
  for (int ks = 0; ks < 4; ++ks) {
    const int d0 = 16 * wv + 4 * ks + 2 * hi;
    v2f a = *reinterpret_cast<const v2f*>(&smem[kVOff + bl * kStride + d0]);
    #pragma unroll
    for (int nt = 0; nt < 4; ++nt) {
      v2f bv = *reinterpret_cast<const v2f*>(&smem[kWOff + (16 * nt + bl) * kStride + d0]);
      acc[nt] = wmma_f32(a, bv, acc[nt]);
    }
  }
  store_partials(kWOff + kWBuf);         // buf1: all its prior readers barriered long ago
  __syncthreads();
  reduce_partials(kWOff + kWBuf);        // lds_v now holds u
  __syncthreads();

  if (t < 16) {
    float sres = 0.0f;
    #pragma unroll 4
    for (int f = 0; f < kF; ++f)
      sres += smem[kVOff + t * kStride + f] * smem[kXOff + t * kStride + f];
    out[b0 + t] = sres;
  }
}

extern "C" void kernel_launch(void* const* d_in, const int* in_sizes, int n_in,
                              void* d_out, int out_size, void* d_ws, size_t ws_size,
                              hipStream_t stream) {
  (void)in_sizes; (void)n_in; (void)d_ws; (void)ws_size; (void)out_size;
  const float* x      = (const float*)d_in[0];
  const float* w0     = (const float*)d_in[1];
  const float* w_mid  = (const float*)d_in[2];
  const float* w_last = (const float*)d_in[3];
  float* out = (float*)d_out;

  const size_t shmem = (size_t)kLdsDw * sizeof(float);   // 78336 B (< 320 KB/WGP)
  // Allow >64KB dynamic LDS (host-side attribute; not a stream op, capture-safe).
  (void)hipFuncSetAttribute(reinterpret_cast<const void*>(mps_chain_kernel),
                            hipFuncAttributeMaxDynamicSharedMemorySize, (int)shmem);

  mps_chain_kernel<<<dim3(kBatch / 16), dim3(128), shmem, stream>>>(
      x, w0, w_mid, w_last, out);
}